// AttentionPooling_15848429322627
// MI455X (gfx1250) — compile-verified
//
#include <hip/hip_runtime.h>
#include <hip/hip_bf16.h>
#include <math.h>

#define NUM_SEGMENTS 16384
#define D_FEAT 128
#define EPS 1e-10f
#define TILE_M 16
#define WAVES_PER_BLOCK 8
#define BLOCK_ROWS (TILE_M * WAVES_PER_BLOCK)   // 128 rows per block
#define LDSW 136   // padded LDS row stride (dwords): 2 rows == 16 mod 64 banks
                   // -> lane halves (rows 4s vs 4s+2) hit disjoint bank sets

typedef float v2f __attribute__((ext_vector_type(2)));
typedef float v8f __attribute__((ext_vector_type(8)));
typedef int   v4i_async __attribute__((vector_size(16)));   // int4 payload type

// ---- monotone unsigned encoding of float for atomicMax-based segment_max ----
__device__ __forceinline__ unsigned enc_f32(float f) {
    unsigned u = __float_as_uint(f);
    return (u & 0x80000000u) ? ~u : (u | 0x80000000u);
}
__device__ __forceinline__ float dec_f32(unsigned e) {
    unsigned u = (e & 0x80000000u) ? (e ^ 0x80000000u) : ~e;
    return __uint_as_float(u);
}

// ---------------------------------------------------------------- init kernel
__global__ void k_init(float* __restrict__ out, float* __restrict__ denom,
                       unsigned* __restrict__ segmax, int out_n) {
    int i = blockIdx.x * blockDim.x + threadIdx.x;
    if (i < out_n) out[i] = 0.0f;
    if (i < NUM_SEGMENTS) { denom[i] = 0.0f; segmax[i] = 0u; }
}

// ------------------------------------------------- gate logits + segment max
// one wave32 per row: 32 lanes x float4 = 128 features, fully coalesced
__global__ void k_gate(const float* __restrict__ x, const int* __restrict__ idx,
                       const float* __restrict__ gw, const float* __restrict__ gb,
                       float* __restrict__ gate, unsigned* __restrict__ segmax,
                       int n) {
    int row  = (blockIdx.x * blockDim.x + threadIdx.x) >> 5;
    int lane = threadIdx.x & 31;
    if (row >= n) return;
    float4 xv = ((const float4*)(x + (size_t)row * D_FEAT))[lane];
    float4 wv = ((const float4*)gw)[lane];
    float d = xv.x * wv.x + xv.y * wv.y + xv.z * wv.z + xv.w * wv.w;
    #pragma unroll
    for (int o = 16; o > 0; o >>= 1) d += __shfl_xor(d, o, 32);
    if (lane == 0) {
        d += gb[0];
        gate[row] = d;
        atomicMax(&segmax[idx[row]], enc_f32(d));
    }
}

// -------------------------------------- exp(gate - segmax) * w^p, denom sum
__global__ void k_expsum(const float* __restrict__ gate, const int* __restrict__ idx,
                         const float* __restrict__ w, const float* __restrict__ powp,
                         const unsigned* __restrict__ segmax,
                         float* __restrict__ gnum, float* __restrict__ denom, int n) {
    int i = blockIdx.x * blockDim.x + threadIdx.x;
    if (i >= n) return;
    int s = idx[i];
    float mx = dec_f32(segmax[s]);
    float g = __expf(gate[i] - mx) * __powf(w[i], powp[0]);
    gnum[i] = g;
    atomicAdd(&denom[s], g);
}

// ------------------------------- message GEMM (WMMA f32) + gated segment-sum
// wave owns a 16-row tile; block stages msg_w (padded, ~68KB) + msg_b into LDS.
__global__ void k_msg_pool(const float* __restrict__ x, const int* __restrict__ idx,
                           const float* __restrict__ msg_w, const float* __restrict__ msg_b,
                           const float* __restrict__ gnum, const float* __restrict__ denom,
                           float* __restrict__ out, int n) {
    __shared__ float lds_w[D_FEAT * LDSW];
    __shared__ float lds_b[D_FEAT];

    int t = threadIdx.x;
    {   // stage msg_w ([K][N] row-major, padded row stride LDSW) and msg_b
#if __has_builtin(__builtin_amdgcn_global_load_async_to_lds_b128)
        // CDNA5 async global->LDS DMA: bypasses VGPRs, tracked by ASYNCcnt
        for (int i = t; i < (D_FEAT * D_FEAT) / 4; i += 256) {
            int k = i >> 5, c = i & 31;                 // row k, float4-chunk c
            v4i_async* g = (v4i_async*)(msg_w + i * 4);
            v4i_async* l = (v4i_async*)&lds_w[k * LDSW + 4 * c];
            __builtin_amdgcn_global_load_async_to_lds_b128(g, l, 0, 0);
        }
#if __has_builtin(__builtin_amdgcn_s_wait_asynccnt)
        __builtin_amdgcn_s_wait_asynccnt(0);
#else
        asm volatile("s_wait_asynccnt 0" ::: "memory");
#endif
#else
        const float4* src = (const float4*)msg_w;
        for (int i = t; i < (D_FEAT * D_FEAT) / 4; i += 256) {
            int k = i >> 5, c = i & 31;
            *(float4*)&lds_w[k * LDSW + 4 * c] = src[i];
        }
#endif
        if (t < D_FEAT) lds_b[t] = msg_b[t];
    }
    __syncthreads();

    int waveId = t >> 5;
    int lane   = t & 31;
    int hi     = lane >> 4;       // lane half: selects K pair (A) / K rows (B)
    int m      = lane & 15;
    int row0   = (blockIdx.x * WAVES_PER_BLOCK + waveId) * TILE_M;
    if (row0 >= n) return;

    // A tile 16x128 in registers: per k-step, lanes 0-15 hold (K0,K1) of row m,
    // lanes 16-31 hold (K2,K3) of row m  (ISA 16x4 f32 A layout)
    v2f a[32];
    {
        int r = row0 + m; if (r >= n) r = n - 1;
        const float* xrow = x + (size_t)r * D_FEAT;
        #pragma unroll
        for (int s = 0; s < 32; ++s)
            a[s] = *(const v2f*)(xrow + 4 * s + 2 * hi);
    }

    // per-row gating scale + segment id; C-tile VGPR j maps to row M = j + 8*hi
    int   seg[8];
    float scl[8];
    #pragma unroll
    for (int j = 0; j < 8; ++j) {
        int r = row0 + j + 8 * hi;
        if (r < n) {
            int s  = idx[r];
            seg[j] = s;
            scl[j] = gnum[r] / (denom[s] + EPS);
        } else { seg[j] = -1; scl[j] = 0.0f; }
    }

    // sorted index => uniform-segment fast path when the whole 16-row tile
    // lands in one segment (single atomic per column instead of 16)
    int  lastr = row0 + TILE_M - 1;
    bool uni   = (lastr < n) && (idx[row0] == idx[lastr]);
    int  useg  = idx[row0];
    float sumscl = 0.0f;
    #pragma unroll
    for (int j = 0; j < 8; ++j) sumscl += scl[j];
    sumscl += __shfl_xor(sumscl, 16, 32);

    #pragma unroll 1
    for (int nt = 0; nt < 8; ++nt) {
        int ncol = nt * 16 + m;
        v8f c = {0.f, 0.f, 0.f, 0.f, 0.f, 0.f, 0.f, 0.f};
        #pragma unroll
        for (int s = 0; s < 32; ++s) {
            // B 4x16: VGPR0 = row (K0 | K2), VGPR1 = row (K1 | K3), col = m
            const float* wk = &lds_w[(4 * s + 2 * hi) * LDSW + ncol];
            v2f b;
            b.x = wk[0];
            b.y = wk[LDSW];
            c = __builtin_amdgcn_wmma_f32_16x16x4_f32(
                    false, a[s], false, b, (short)0, c, false, false);
        }
        float bias = lds_b[ncol];
        if (uni) {
            float v = 0.0f;
            #pragma unroll
            for (int j = 0; j < 8; ++j) v += c[j] * scl[j];
            v += __shfl_xor(v, 16, 32);           // reduce across row halves
            if (hi == 0)
                atomicAdd(&out[(size_t)useg * D_FEAT + ncol], v + bias * sumscl);
        } else {
            #pragma unroll
            for (int j = 0; j < 8; ++j)
                if (seg[j] >= 0)
                    atomicAdd(&out[(size_t)seg[j] * D_FEAT + ncol],
                              scl[j] * (c[j] + bias));
        }
    }
}

// ---------------------------------------------------------------------------
extern "C" void kernel_launch(void* const* d_in, const int* in_sizes, int n_in,
                              void* d_out, int out_size, void* d_ws, size_t ws_size,
                              hipStream_t stream) {
    const float* x       = (const float*)d_in[0];
    const int*   index   = (const int*)  d_in[1];
    const float* weights = (const float*)d_in[2];
    const float* gate_w  = (const float*)d_in[3];
    const float* gate_b  = (const float*)d_in[4];
    const float* msg_w   = (const float*)d_in[5];
    const float* msg_b   = (const float*)d_in[6];
    const float* pow_p   = (const float*)d_in[7];
    float*       out     = (float*)d_out;
    const int n = in_sizes[1];   // number of rows (== len(index))

    // workspace layout
    float*    gate   = (float*)d_ws;                       // n
    float*    gnum   = gate + n;                           // n
    float*    denom  = gnum + n;                           // NUM_SEGMENTS
    unsigned* segmax = (unsigned*)(denom + NUM_SEGMENTS);  // NUM_SEGMENTS

    k_init<<<(out_size + 255) / 256, 256, 0, stream>>>(out, denom, segmax, out_size);

    // wave-per-row: 8 rows per 256-thread block
    k_gate<<<(n + 7) / 8, 256, 0, stream>>>(x, index, gate_w, gate_b,
                                            gate, segmax, n);

    k_expsum<<<(n + 255) / 256, 256, 0, stream>>>(gate, index, weights, pow_p,
                                                  segmax, gnum, denom, n);

    k_msg_pool<<<(n + BLOCK_ROWS - 1) / BLOCK_ROWS, 256, 0, stream>>>(
        x, index, msg_w, msg_b, gnum, denom, out, n);
}